// SkipGram_55396488184470
// MI455X (gfx1250) — compile-verified
//
#include <hip/hip_runtime.h>

// SkipGram negative-sampling loss for MI455X (gfx1250, wave32).
//
// Roofline: 96 MB of embedding gathers vs 46 MFLOP -> memory bound
// (~4 us at 23.3 TB/s). WMMA f32 16x16x4 computes full 16x16 F*G^T tiles
// (we need only the diagonal) because the redundant FLOPs are free and the
// data movement stays exactly minimal: each gathered row is loaded once
// per use, A (focus) fragments are register-resident across all 11 GEMMs.

typedef __attribute__((ext_vector_type(2))) float v2f;
typedef __attribute__((ext_vector_type(8))) float v8f;

#define EMB_DIM 128
#define NEG_K 10

// C/D 16x16 f32 layout: VGPR v, lanes 0-15 -> (M=v, N=lane);
// lanes 16-31 -> (M=v+8, N=lane-16).
// Diagonal (m,m): m<8 -> VGPR m @ lane m;  m>=8 -> VGPR m-8 @ lane m+16.
__device__ __forceinline__ float diag_pick(v8f c, int ci) {
  float d = c[0];
  d = (ci == 1) ? c[1] : d;
  d = (ci == 2) ? c[2] : d;
  d = (ci == 3) ? c[3] : d;
  d = (ci == 4) ? c[4] : d;
  d = (ci == 5) ? c[5] : d;
  d = (ci == 6) ? c[6] : d;
  d = (ci == 7) ? c[7] : d;
  return d;
}

__device__ __forceinline__ float log_sigmoid(float x) {
  // numerically stable: min(x,0) - log1p(exp(-|x|))
  return fminf(x, 0.0f) - log1pf(expf(-fabsf(x)));
}

__global__ void __launch_bounds__(256)
sg_loss_kernel(const float* __restrict__ focus_W,
               const float* __restrict__ context_W,
               const int* __restrict__ focus_idx,
               const int* __restrict__ context_idx,
               const int* __restrict__ neg_idx,
               float* __restrict__ partial,
               int batch) {
  const int lane = threadIdx.x & 31;
  const int wave = threadIdx.x >> 5;
  const int tile = blockIdx.x * (blockDim.x >> 5) + wave;
  const int row0 = tile * 16;
  if (row0 >= batch) return;  // wave-uniform; EXEC stays all-ones for WMMA

  // A-matrix 16x4 f32 fragment layout: lane L holds row (L&15),
  // K pair base = k0 + (L>>4)*2. B (= G^T) mirrors with N on lanes, so a
  // lane loads the SAME float2 offsets from its gathered G row.
  const int sub  = lane & 15;
  const int koff = (lane >> 4) << 1;
  const int row  = row0 + sub;

  const float* Fp = focus_W + (size_t)focus_idx[row] * EMB_DIM + koff;

  // Preload focus-row fragments once (64 VGPRs), reused by all 11 GEMMs.
  v2f afrag[32];
#pragma unroll
  for (int s = 0; s < 32; ++s)
    afrag[s] = *(const v2f*)(Fp + 4 * s);

  const bool active = (lane < 8) || (lane >= 24);
  const int ci = active ? ((lane < 8) ? lane : (lane - 24)) : 0;

  // ---- positive GEMM: 16x16x128 via 32 x V_WMMA_F32_16X16X4_F32 ----
  const float* Cp = context_W + (size_t)context_idx[row] * EMB_DIM + koff;
  v8f c = {};
#pragma unroll
  for (int s = 0; s < 32; ++s) {
    v2f b = *(const v2f*)(Cp + 4 * s);
    c = __builtin_amdgcn_wmma_f32_16x16x4_f32(false, afrag[s], false, b,
                                              (short)0, c, false, false);
  }
  float pos = diag_pick(c, ci);

  // ---- 10 negative GEMMs, accumulate diagonals ----
  float negsum = 0.0f;
  for (int k = 0; k < NEG_K; ++k) {
    const float* Np =
        context_W + (size_t)neg_idx[row * NEG_K + k] * EMB_DIM + koff;
    v8f cn = {};
#pragma unroll
    for (int s = 0; s < 32; ++s) {
      v2f b = *(const v2f*)(Np + 4 * s);
      cn = __builtin_amdgcn_wmma_f32_16x16x4_f32(false, afrag[s], false, b,
                                                 (short)0, cn, false, false);
    }
    negsum += diag_pick(cn, ci);
  }

  // Active lanes {0..7, 24..31} carry rows m = lane&15 of this tile.
  float lp = log_sigmoid(pos);
  float ln = log_sigmoid(negsum);
  float contrib =
      active ? ((1.0f - lp) * (1.0f - lp) + ln * ln) : 0.0f;

  // wave32 reduction
#pragma unroll
  for (int off = 16; off >= 1; off >>= 1)
    contrib += __shfl_down(contrib, off, 32);
  if (lane == 0) partial[tile] = contrib;
}

// Deterministic final reduction of per-tile partials (fixed-shape tree).
__global__ void __launch_bounds__(256)
sg_reduce_kernel(const float* __restrict__ partial, float* __restrict__ out,
                 int n) {
  __shared__ float s[256];
  float v = 0.0f;
  for (int i = threadIdx.x; i < n; i += 256) v += partial[i];
  s[threadIdx.x] = v;
  __syncthreads();
  for (int st = 128; st > 0; st >>= 1) {
    if ((int)threadIdx.x < st) s[threadIdx.x] += s[threadIdx.x + st];
    __syncthreads();
  }
  if (threadIdx.x == 0) out[0] = s[0];
}

extern "C" void kernel_launch(void* const* d_in, const int* in_sizes, int n_in,
                              void* d_out, int out_size, void* d_ws,
                              size_t ws_size, hipStream_t stream) {
  const float* focus_W   = (const float*)d_in[0];
  const float* context_W = (const float*)d_in[1];
  const int* focus_idx   = (const int*)d_in[2];
  const int* context_idx = (const int*)d_in[3];
  const int* neg_idx     = (const int*)d_in[4];
  float* out = (float*)d_out;

  const int batch = in_sizes[2];          // 16384
  const int tiles = (batch + 15) / 16;    // 1024 tiles of 16 rows
  float* partial = (float*)d_ws;          // tiles * 4 bytes of scratch

  const int waves_per_block = 8;          // 256 threads = 8 wave32
  const int blocks = (tiles + waves_per_block - 1) / waves_per_block;

  sg_loss_kernel<<<blocks, waves_per_block * 32, 0, stream>>>(
      focus_W, context_W, focus_idx, context_idx, neg_idx, partial, batch);
  sg_reduce_kernel<<<1, 256, 0, stream>>>(partial, out, tiles);
}